// MergedQKVParallelLinearWithLora_43628277793193
// MI455X (gfx1250) — compile-verified
//
#include <hip/hip_runtime.h>

typedef __attribute__((ext_vector_type(2))) float v2f;
typedef __attribute__((ext_vector_type(8))) float v8f;

#define D_DIM 2048
#define QS_SZ 2048
#define KVS_SZ 512
#define O_TOT 3072
#define T_TOK 32768
#define L_MAX 8
#define R_RANK 16
#define LRK 128          // L_MAX * R_RANK, expand-GEMM K
#define SLR 384          // S row length = 3 * 128
#define BM 128
#define BN 128
#define BK 32
#define LDB 36           // padded row length (36*4=144B keeps 16B chunks aligned,
                         // 36*m mod 64 distinct for m in [0,16) -> conflict-free)
#define NKC1 (D_DIM / BK)   // 64 base-GEMM chunks
#define NKC2 (LRK / BK)     // 4 expand chunks

#if defined(__gfx1250__)
#define HAVE_ASYNC 1
#else
#define HAVE_ASYNC 0
#endif

__device__ __forceinline__ void asyncCopy16(const float* g, float* l) {
#if HAVE_ASYNC
  // VDST = 32-bit LDS byte address (low 32 bits of the generic LDS pointer,
  // per the aperture rule); VADDR = 64-bit global addr.
  unsigned ldsOff = (unsigned)(unsigned long long)l;
  unsigned long long gaddr = (unsigned long long)g;
  asm volatile("global_load_async_to_lds_b128 %0, %1, off"
               :
               : "v"(ldsOff), "v"(gaddr)
               : "memory");
#else
  *(float4*)l = *(const float4*)g;
#endif
}

__device__ __forceinline__ void waitCopies() {
#if HAVE_ASYNC
#if __has_builtin(__builtin_amdgcn_s_wait_asynccnt)
  __builtin_amdgcn_s_wait_asynccnt(0);
#else
  asm volatile("s_wait_asynccnt 0x0" ::: "memory");
#endif
#endif
  __syncthreads();
}

// Load a 128x32 row-major tile src[r*rowStride + k] -> lds[r*LDB + k].
__device__ __forceinline__ void loadTile(float* lds, const float* src,
                                         int rowStride, int tid) {
#pragma unroll
  for (int i = 0; i < 4; ++i) {
    int f = tid + i * 256;
    int r = f >> 3;            // 8 float4 per row
    int k = (f & 7) << 2;
    asyncCopy16(src + (size_t)r * rowStride + k, lds + r * LDB + k);
  }
}

// One BK=32 K-chunk of 16x16x4 f32 WMMAs. Wave covers 32(M) x 64(N): 2x4 tiles.
__device__ __forceinline__ void mmaChunk(const float* xs, const float* bs,
                                         v8f acc[2][4], int lane, int waveM,
                                         int waveN) {
#pragma unroll
  for (int k = 0; k < BK; k += 4) {
    int kA = k + ((lane >> 4) << 1);  // lanes 16-31 hold K=2,3
    int ml = lane & 15;
    v2f a[2], b[4];
#pragma unroll
    for (int mt = 0; mt < 2; ++mt) {
      int m = waveM * 32 + mt * 16 + ml;
      a[mt] = *(const v2f*)(xs + m * LDB + kA);   // single ds_load_b64
    }
#pragma unroll
    for (int nt = 0; nt < 4; ++nt) {
      int n = waveN * 64 + nt * 16 + ml;
      b[nt] = *(const v2f*)(bs + n * LDB + kA);
    }
#pragma unroll
    for (int mt = 0; mt < 2; ++mt)
#pragma unroll
      for (int nt = 0; nt < 4; ++nt)
        acc[mt][nt] = __builtin_amdgcn_wmma_f32_16x16x4_f32(
            false, a[mt], false, b[nt], (short)0, acc[mt][nt], false, false);
  }
}

// Kernel 1: masked shrink. S[t, slice, l*16+r] = (idx[t]==l) ? x[t,:]·A[l,r,:] : 0
__global__ __launch_bounds__(256) void lora_shrink_kernel(
    const float* __restrict__ x, const float* __restrict__ Aq,
    const float* __restrict__ Ak, const float* __restrict__ Av,
    const int* __restrict__ tli, float* __restrict__ S) {
  __shared__ float xs[2][BM * LDB];
  __shared__ float as[2][BM * LDB];
  __shared__ int idx_s[BM];
  int tid = threadIdx.x, lane = tid & 31, wid = tid >> 5;
  int waveM = wid & 3, waveN = wid >> 2;
  int t0 = blockIdx.x * BM;
  int slice = blockIdx.y;
  const float* A = (slice == 0) ? Aq : ((slice == 1) ? Ak : Av);

  auto issue = [&](int c, int buf) {
    int k0 = c * BK;
    loadTile(xs[buf], x + (size_t)t0 * D_DIM + k0, D_DIM, tid);
    loadTile(as[buf], A + k0, D_DIM, tid);  // 128 rows (l*16+r), stride D
  };

  v8f acc[2][4];
#pragma unroll
  for (int mt = 0; mt < 2; ++mt)
#pragma unroll
    for (int nt = 0; nt < 4; ++nt)
#pragma unroll
      for (int j = 0; j < 8; ++j) acc[mt][nt][j] = 0.f;

  issue(0, 0);
  if (tid < BM) idx_s[tid] = tli[t0 + tid];
  waitCopies();
  for (int c = 0; c < NKC1; ++c) {
    int cur = c & 1;
    if (c + 1 < NKC1) issue(c + 1, cur ^ 1);
    mmaChunk(xs[cur], as[cur], acc, lane, waveM, waveN);
    waitCopies();
  }

#pragma unroll
  for (int mt = 0; mt < 2; ++mt) {
#pragma unroll
    for (int nt = 0; nt < 4; ++nt) {
      int nl = waveN * 64 + nt * 16 + (lane & 15);
      int l = nl >> 4;
#pragma unroll
      for (int i = 0; i < 8; ++i) {
        int ml = waveM * 32 + mt * 16 + i + ((lane >> 4) << 3);
        float v = (idx_s[ml] == l) ? acc[mt][nt][i] : 0.f;
        S[(size_t)(t0 + ml) * SLR + slice * 128 + nl] = v;
      }
    }
  }
}

// Kernel 2: out = x@Wqkv^T + bias + S_masked@Bcat^T + gated lora bias
__global__ __launch_bounds__(256) void qkv_lora_kernel(
    const float* __restrict__ x, const float* __restrict__ Wqkv,
    const float* __restrict__ bias, const float* __restrict__ Bq,
    const float* __restrict__ Bk, const float* __restrict__ Bv,
    const float* __restrict__ lbq, const float* __restrict__ lbk,
    const float* __restrict__ lbv, const int* __restrict__ tli,
    const float* __restrict__ S, float* __restrict__ out) {
  __shared__ float xs[2][BM * LDB];
  __shared__ float bs[2][BM * LDB];
  __shared__ int idx_s[BM];
  int tid = threadIdx.x, lane = tid & 31, wid = tid >> 5;
  int waveM = wid & 3, waveN = wid >> 2;
  int t0 = blockIdx.x * BM;
  int by = blockIdx.y;
  int o0 = by * BN;
  int slice = (by < 16) ? 0 : ((by < 20) ? 1 : 2);
  int sliceBase = (slice == 0) ? 0 : ((slice == 1) ? QS_SZ : QS_SZ + KVS_SZ);
  int OS = (slice == 0) ? QS_SZ : KVS_SZ;
  const float* B = (slice == 0) ? Bq : ((slice == 1) ? Bk : Bv);
  const float* lb = (slice == 0) ? lbq : ((slice == 1) ? lbk : lbv);
  int oo0 = o0 - sliceBase;  // column offset within this slice

  auto issue = [&](int c, int buf) {
    if (c < NKC1) {
      int k0 = c * BK;
      loadTile(xs[buf], x + (size_t)t0 * D_DIM + k0, D_DIM, tid);
      loadTile(bs[buf], Wqkv + (size_t)o0 * D_DIM + k0, D_DIM, tid);
    } else {
      int kk0 = (c - NKC1) * BK;
      loadTile(xs[buf], S + (size_t)t0 * SLR + slice * 128 + kk0, SLR, tid);
#pragma unroll
      for (int i = 0; i < 4; ++i) {  // Bcat tile: B[(kk)/16, oo0+n, (kk)%16]
        int f = tid + i * 256;
        int n = f >> 3;
        int k = (f & 7) << 2;
        int l = (kk0 + k) >> 4;
        int r = (kk0 + k) & 15;
        asyncCopy16(B + ((size_t)(l * OS + oo0 + n)) * R_RANK + r,
                    bs[buf] + n * LDB + k);
      }
    }
  };

  v8f acc[2][4];
#pragma unroll
  for (int mt = 0; mt < 2; ++mt)
#pragma unroll
    for (int nt = 0; nt < 4; ++nt)
#pragma unroll
      for (int j = 0; j < 8; ++j) acc[mt][nt][j] = 0.f;

  issue(0, 0);
  if (tid < BM) idx_s[tid] = tli[t0 + tid];
  waitCopies();
  const int NC = NKC1 + NKC2;  // 64 base chunks + 4 lora-expand chunks
  for (int c = 0; c < NC; ++c) {
    int cur = c & 1;
    if (c + 1 < NC) issue(c + 1, cur ^ 1);
    mmaChunk(xs[cur], bs[cur], acc, lane, waveM, waveN);
    waitCopies();
  }

  // Epilogue: biases and store
#pragma unroll
  for (int mt = 0; mt < 2; ++mt) {
#pragma unroll
    for (int nt = 0; nt < 4; ++nt) {
      int nl = waveN * 64 + nt * 16 + (lane & 15);
#pragma unroll
      for (int i = 0; i < 8; ++i) {
        int ml = waveM * 32 + mt * 16 + i + ((lane >> 4) << 3);
        int li = idx_s[ml];
        float v = acc[mt][nt][i] + bias[o0 + nl];
        if (li >= 0) v += lb[(size_t)li * OS + oo0 + nl];
        out[(size_t)(t0 + ml) * O_TOT + o0 + nl] = v;
      }
    }
  }
}

extern "C" void kernel_launch(void* const* d_in, const int* in_sizes, int n_in,
                              void* d_out, int out_size, void* d_ws,
                              size_t ws_size, hipStream_t stream) {
  const float* x = (const float*)d_in[0];
  const float* Wqkv = (const float*)d_in[1];
  const float* bias = (const float*)d_in[2];
  const float* Aq = (const float*)d_in[3];
  const float* Ak = (const float*)d_in[4];
  const float* Av = (const float*)d_in[5];
  const float* Bq = (const float*)d_in[6];
  const float* Bk = (const float*)d_in[7];
  const float* Bv = (const float*)d_in[8];
  const float* lbq = (const float*)d_in[9];
  const float* lbk = (const float*)d_in[10];
  const float* lbv = (const float*)d_in[11];
  const int* tli = (const int*)d_in[12];
  float* out = (float*)d_out;
  float* S = (float*)d_ws;  // (T, 3, 128) masked shrink results

  dim3 blk(256);
  dim3 g1(T_TOK / BM, 3);
  lora_shrink_kernel<<<g1, blk, 0, stream>>>(x, Aq, Ak, Av, tli, S);

  dim3 g2(T_TOK / BM, O_TOT / BN);
  qkv_lora_kernel<<<g2, blk, 0, stream>>>(x, Wqkv, bias, Bq, Bk, Bv, lbq, lbk,
                                          lbv, tli, S, out);
}